// CCELoss_19292993094305
// MI455X (gfx1250) — compile-verified
//
#include <hip/hip_runtime.h>
#include <hip/hip_bf16.h>

// CCELoss (calibration error) for MI455X / gfx1250.
//
// Roofline: 167 MB read once @ 23.3 TB/s => ~7.2 us floor; compute is
// negligible. The limiter beyond HBM is the LDS atomic pipe: most softmax
// probabilities fall into bin 0 of their class, so same-address ds_add_f32
// conflicts would serialize up to 32x per wave. We replicate the LDS
// histogram 16x (lane-indexed) so worst-case conflict degree is 2, and the
// replica stride (570 floats, 58 mod 64 banks) spreads the 16 copies of one
// bin across 16 distinct banks. Logits stream via non-temporal b128 loads
// (read-once data; don't pollute the 192 MB L2).
//
// WMMA (V_WMMA_F32_16X16X4_F32) is used in the finalize reduction: with an
// all-ones A matrix, D[m][n] = sum_k B[k][n] + C[m][n]; a total sum is
// packing-order invariant, so the 190 bin values are packed arbitrarily
// into three 4x16 B chunks and accumulated through C in full f32.

#define N_CLASSES 19
#define N_BINS    10
#define NSEG      (N_CLASSES * N_BINS)   // 190
#define NREP      16                     // LDS histogram replicas
#define EPSF      1e-13f

typedef __attribute__((ext_vector_type(2))) float v2f;
typedef __attribute__((ext_vector_type(4))) float v4f;
typedef __attribute__((ext_vector_type(4))) int   v4i;
typedef __attribute__((ext_vector_type(8))) float v8f;

// ---------------------------------------------------------------------------
// Zero the global accumulator workspace (3 * 190 floats).
// ---------------------------------------------------------------------------
__global__ void cce_zero_ws_kernel(float* __restrict__ ws, int n) {
    int i = blockIdx.x * blockDim.x + threadIdx.x;
    if (i < n) ws[i] = 0.0f;
}

// ---------------------------------------------------------------------------
// Per-pixel softmax + binning into a replica of the block-private LDS
// histogram (ds_add_f32), flushed with global_atomic_add_f32.
// h points at this lane's replica: [conf(190) | cnt(190) | acc(190)].
// ---------------------------------------------------------------------------
__device__ __forceinline__ void process_pixel(const float xv[N_CLASSES], int tgt,
                                              float* __restrict__ h) {
    float m = xv[0];
#pragma unroll
    for (int c = 1; c < N_CLASSES; ++c) m = fmaxf(m, xv[c]);

    float e[N_CLASSES];
    float s = 0.0f;
#pragma unroll
    for (int c = 0; c < N_CLASSES; ++c) {
        e[c] = __expf(xv[c] - m);
        s += e[c];
    }
    float inv = 1.0f / s;  // s >= 1 (max-subtracted), always finite

#pragma unroll
    for (int c = 0; c < N_CLASSES; ++c) {
        float p = e[c] * inv;
        if (p > 0.0f) {  // p == 0 belongs to no bin (matches reference)
            int b = (int)ceilf(p * (float)N_BINS) - 1;
            b = b < 0 ? 0 : (b > N_BINS - 1 ? N_BINS - 1 : b);
            int seg = c * N_BINS + b;
            atomicAdd(&h[seg], p);                 // conf_sum
            atomicAdd(&h[NSEG + seg], 1.0f);       // no_pred
            if (c == tgt) atomicAdd(&h[2 * NSEG + seg], 1.0f);  // no_acc
        }
    }
}

__global__ __launch_bounds__(256) void cce_hist_kernel(
    const float* __restrict__ logits,   // [C, n_pix] (B=1)
    const int*   __restrict__ target,   // [n_pix]
    float*       __restrict__ ws,       // [3*NSEG] global accumulators
    int n_pix) {
    // 16 replicas x [conf|cnt|acc] = 16 * 570 floats = 36.5 KB (of 320 KB).
    __shared__ float s_hist[NREP * 3 * NSEG];

    for (int i = threadIdx.x; i < NREP * 3 * NSEG; i += blockDim.x)
        s_hist[i] = 0.0f;
    __syncthreads();

    float* h = s_hist + (threadIdx.x & (NREP - 1)) * (3 * NSEG);

    // Four pixels per thread: each class plane is a contiguous non-temporal
    // global_load_b128 stream per wave (full coalescing, no L2 pollution).
    int pix = (blockIdx.x * blockDim.x + threadIdx.x) * 4;
    if (pix + 3 < n_pix) {
        v4f x[N_CLASSES];
#pragma unroll
        for (int c = 0; c < N_CLASSES; ++c)
            x[c] = __builtin_nontemporal_load(
                reinterpret_cast<const v4f*>(logits + (size_t)c * n_pix + pix));
        v4i t4 = __builtin_nontemporal_load(
            reinterpret_cast<const v4i*>(target + pix));

#pragma unroll
        for (int comp = 0; comp < 4; ++comp) {
            float xv[N_CLASSES];
#pragma unroll
            for (int c = 0; c < N_CLASSES; ++c) xv[c] = x[c][comp];
            process_pixel(xv, t4[comp], h);
        }
    }

    __syncthreads();
    // Reduce the 16 replicas, then one global f32 atomic per bin per block.
    for (int i = threadIdx.x; i < 3 * NSEG; i += blockDim.x) {
        float acc = 0.0f;
#pragma unroll
        for (int r = 0; r < NREP; ++r) acc += s_hist[r * (3 * NSEG) + i];
        atomicAdd(&ws[i], acc);   // global_atomic_add_f32
    }
}

// ---------------------------------------------------------------------------
// Finalize: one wave32, WMMA-based sum reductions (see header comment).
// Each lane's d[0] is one column sum of the accumulated B chunks; lanes
// 16-31 duplicate lanes 0-15 (D rows are identical with a ones A), so the
// 32-lane sum equals 2x the total.
// ---------------------------------------------------------------------------
__device__ __forceinline__ float wave_sum32(float v) {
#pragma unroll
    for (int off = 16; off >= 1; off >>= 1) v += __shfl_xor(v, off, 32);
    return v;
}

__global__ __launch_bounds__(32) void cce_finalize_kernel(
    const float* __restrict__ ws, float* __restrict__ out) {
    const float* conf = ws;
    const float* cnt  = ws + NSEG;
    const float* acc  = ws + 2 * NSEG;
    int lane = threadIdx.x & 31;

    v2f ones;
    ones[0] = 1.0f;
    ones[1] = 1.0f;

    // Pass 1: total = sum of all bin counts.
    v8f d = {0.f, 0.f, 0.f, 0.f, 0.f, 0.f, 0.f, 0.f};
#pragma unroll
    for (int j = 0; j < 3; ++j) {
        int i0 = j * 64 + lane * 2;
        v2f b;
        b[0] = (i0     < NSEG) ? cnt[i0]     : 0.0f;
        b[1] = (i0 + 1 < NSEG) ? cnt[i0 + 1] : 0.0f;
        d = __builtin_amdgcn_wmma_f32_16x16x4_f32(
            /*neg_a=*/false, ones, /*neg_b=*/false, b,
            /*c_mod=*/(short)0, d, /*reuse_a=*/false, /*reuse_b=*/false);
    }
    float total = 0.5f * wave_sum32(d[0]);
    float inv_total = 1.0f / total;

    // Pass 2: loss = sum_bins ((acc-conf)/(cnt+eps))^2 * cnt / total.
    v8f d2 = {0.f, 0.f, 0.f, 0.f, 0.f, 0.f, 0.f, 0.f};
#pragma unroll
    for (int j = 0; j < 3; ++j) {
        int i0 = j * 64 + lane * 2;
        v2f b;
#pragma unroll
        for (int s = 0; s < 2; ++s) {
            int i = i0 + s;
            float term = 0.0f;
            if (i < NSEG) {
                float n    = cnt[i];
                float diff = (acc[i] - conf[i]) / (n + EPSF);
                term = diff * diff * n * inv_total;
            }
            b[s] = term;
        }
        d2 = __builtin_amdgcn_wmma_f32_16x16x4_f32(
            false, ones, false, b, (short)0, d2, false, false);
    }
    float loss = 0.5f * wave_sum32(d2[0]);
    if (threadIdx.x == 0) out[0] = loss;
}

// ---------------------------------------------------------------------------
extern "C" void kernel_launch(void* const* d_in, const int* in_sizes, int n_in,
                              void* d_out, int out_size, void* d_ws, size_t ws_size,
                              hipStream_t stream) {
    const float* logits = (const float*)d_in[0];  // [1,19,1024,2048] f32
    const int*   target = (const int*)d_in[1];    // [1,1024,2048] i32
    float* ws  = (float*)d_ws;
    float* out = (float*)d_out;

    int n_pix = in_sizes[1];  // 1024*2048 = 2,097,152

    cce_zero_ws_kernel<<<(3 * NSEG + 255) / 256, 256, 0, stream>>>(ws, 3 * NSEG);

    const int threads = 256;              // 8 wave32s per block
    int work   = n_pix / 4;               // 4 pixels per thread
    int blocks = (work + threads - 1) / threads;
    cce_hist_kernel<<<blocks, threads, 0, stream>>>(logits, target, ws, n_pix);

    cce_finalize_kernel<<<1, 32, 0, stream>>>(ws, out);
}